// ViSNet_P3M_18081812316182
// MI455X (gfx1250) — compile-verified
//
#include <hip/hip_runtime.h>

// ViSNet layer for gfx1250. f32 WMMA (16x16x4) everywhere; edge path fully fused.
// edge_index treated as int32 (JAX default x64-disabled).
// Streaming [E,H] tensors (f_ij reads, df_ij writes) use non-temporal hints so the
// 192MB L2 retains the gathered node tensors (qkv, vec, P2, aggregators).

#define DEV __device__ __forceinline__

typedef float v2f __attribute__((ext_vector_type(2)));
typedef float v8f __attribute__((ext_vector_type(8)));

constexpr int NN = 10000;
constexpr int EE = 320000;

DEV v8f wmma4(v2f a, v2f b, v8f c) {
  // D = A(16x4, f32) * B(4x16, f32) + C(16x16, f32)
  return __builtin_amdgcn_wmma_f32_16x16x4_f32(false, a, false, b, (short)0, c, false, false);
}

DEV float silu_f(float x) { return x / (1.0f + __expf(-x)); }
DEV float cutoff_f(float r) {
  return (r < 5.0f) ? 0.5f * (__cosf(r * 0.62831853071795864769f) + 1.0f) : 0.0f;
}

DEV v8f splat8(float b) {
  v8f c;
#pragma unroll
  for (int i = 0; i < 8; ++i) c[i] = b;
  return c;
}

// Pack weights into LDS as k-pairs: wl[kk*ldw + coloff + n] = (W[2kk][n], W[2kk+1][n])
DEV void load_w(v2f* wl, int ldw, int coloff, const float* __restrict__ W, int ncols,
                int tid, int nthr) {
  int total = 64 * ncols;
  for (int i = tid; i < total; i += nthr) {
    int kk = i / ncols, n = i - kk * ncols;
    v2f t;
    t.x = W[(2 * kk) * ncols + n];
    t.y = W[(2 * kk + 1) * ncols + n];
    wl[kk * ldw + coloff + n] = t;
  }
}

// A fragments for one 16-row tile, K=128: a[s] covers k = 4s..4s+3 (split by half-wave)
DEV void load_a32(const float* __restrict__ rowp, int hf, v2f* a) {
#pragma unroll
  for (int s = 0; s < 32; ++s) a[s] = *(const v2f*)(rowp + 4 * s + 2 * hf);
}

// Non-temporal variant for streamed-once tensors (f_ij)
DEV void load_a32_nt(const float* __restrict__ rowp, int hf, v2f* a) {
#pragma unroll
  for (int s = 0; s < 32; ++s)
    a[s] = __builtin_nontemporal_load((const v2f*)(rowp + 4 * s + 2 * hf));
}

DEV void gemm_pair(const v2f* a, const v2f* wl, int ldw, int col0, int col1,
                   int l16, int hf, v8f& c0, v8f& c1) {
#pragma unroll
  for (int s = 0; s < 32; ++s) {
    v2f b0 = wl[(2 * s + hf) * ldw + col0 + l16];
    v2f b1 = wl[(2 * s + hf) * ldw + col1 + l16];
    c0 = wmma4(a[s], b0, c0);
    c1 = wmma4(a[s], b1, c1);
  }
}

DEV void gemm_one(const v2f* a, const v2f* wl, int ldw, int col, int l16, int hf, v8f& c0) {
#pragma unroll
  for (int s = 0; s < 32; ++s) {
    v2f b0 = wl[(2 * s + hf) * ldw + col + l16];
    c0 = wmma4(a[s], b0, c0);
  }
}

// ---------------- Kernel 1: qkv = x @ [Wq|Wk|Wv] + [bq|bk|bv]  -> ws [N][384]
__global__ __launch_bounds__(256) void k_qkv(
    const float* __restrict__ x,
    const float* __restrict__ Wq, const float* __restrict__ bq,
    const float* __restrict__ Wk, const float* __restrict__ bk,
    const float* __restrict__ Wv, const float* __restrict__ bv,
    float* __restrict__ qkv) {
  __shared__ v2f wl[64 * 384];
  int tid = threadIdx.x;
  load_w(wl, 384, 0, Wq, 128, tid, blockDim.x);
  load_w(wl, 384, 128, Wk, 128, tid, blockDim.x);
  load_w(wl, 384, 256, Wv, 128, tid, blockDim.x);
  __syncthreads();
  int lane = tid & 31, l16 = lane & 15, hf = lane >> 4;
  int nw = (gridDim.x * blockDim.x) >> 5;
  for (int job = (blockIdx.x * blockDim.x + tid) >> 5; job < NN / 16; job += nw) {
    int mbase = job * 16;
    v2f a[32];
    load_a32(x + (size_t)(mbase + l16) * 128, hf, a);
    for (int ct = 0; ct < 24; ct += 2) {
      int c0 = ct * 16 + l16, c1 = c0 + 16;
      float b0 = c0 < 128 ? bq[c0] : (c0 < 256 ? bk[c0 - 128] : bv[c0 - 256]);
      float b1 = c1 < 128 ? bq[c1] : (c1 < 256 ? bk[c1 - 128] : bv[c1 - 256]);
      v8f acc0 = splat8(b0), acc1 = splat8(b1);
      gemm_pair(a, wl, 384, ct * 16, ct * 16 + 16, l16, hf, acc0, acc1);
#pragma unroll
      for (int v = 0; v < 8; ++v) {
        size_t r = (size_t)(mbase + v + 8 * hf) * 384;
        qkv[r + c0] = acc0[v];
        qkv[r + c1] = acc1[v];
      }
    }
  }
}

// ---------------- Kernel 2: vec @ Wvec. vec1*vec2 reduced over spatial dim into
// vec_dot (atomics); vec3 stored. Rows = flattened [N*3].
__global__ __launch_bounds__(256) void k_vecproj(
    const float* __restrict__ vec, const float* __restrict__ Wvec,
    float* __restrict__ vec3w, float* __restrict__ vdotw) {
  __shared__ v2f wl[64 * 384];
  int tid = threadIdx.x;
  load_w(wl, 384, 0, Wvec, 384, tid, blockDim.x);
  __syncthreads();
  int lane = tid & 31, l16 = lane & 15, hf = lane >> 4;
  int nw = (gridDim.x * blockDim.x) >> 5;
  for (int job = (blockIdx.x * blockDim.x + tid) >> 5; job < (3 * NN) / 16; job += nw) {
    int mbase = job * 16;
    v2f a[32];
    load_a32(vec + (size_t)(mbase + l16) * 128, hf, a);
    for (int ct = 0; ct < 8; ++ct) {  // vec1 (col ct) and vec2 (col ct+8) together
      int c = ct * 16 + l16;
      v8f p1 = splat8(0.f), p2 = splat8(0.f);
      gemm_pair(a, wl, 384, ct * 16, 128 + ct * 16, l16, hf, p1, p2);
#pragma unroll
      for (int v = 0; v < 8; ++v) {
        int row = mbase + v + 8 * hf;
        atomicAdd(vdotw + (size_t)(row / 3) * 128 + c, p1[v] * p2[v]);
      }
    }
    for (int ct = 0; ct < 8; ct += 2) {  // vec3
      int c = ct * 16 + l16;
      v8f q1 = splat8(0.f), q2 = splat8(0.f);
      gemm_pair(a, wl, 384, 256 + ct * 16, 256 + ct * 16 + 16, l16, hf, q1, q2);
#pragma unroll
      for (int v = 0; v < 8; ++v) {
        size_t r = (size_t)(mbase + v + 8 * hf) * 128;
        vec3w[r + c] = q1[v];
        vec3w[r + c + 16] = q2[v];
      }
    }
  }
}

// ---------------- Kernel 3: P2 = vec @ [Wwtrg|Wwsrc|Wttrg|Wtsrc] -> ws [N*3][512]
__global__ __launch_bounds__(256) void k_vecproj2(
    const float* __restrict__ vec,
    const float* __restrict__ Wwtrg, const float* __restrict__ Wwsrc,
    const float* __restrict__ Wttrg, const float* __restrict__ Wtsrc,
    float* __restrict__ P2) {
  __shared__ v2f wl[64 * 512];
  int tid = threadIdx.x;
  load_w(wl, 512, 0, Wwtrg, 128, tid, blockDim.x);
  load_w(wl, 512, 128, Wwsrc, 128, tid, blockDim.x);
  load_w(wl, 512, 256, Wttrg, 128, tid, blockDim.x);
  load_w(wl, 512, 384, Wtsrc, 128, tid, blockDim.x);
  __syncthreads();
  int lane = tid & 31, l16 = lane & 15, hf = lane >> 4;
  int nw = (gridDim.x * blockDim.x) >> 5;
  for (int job = (blockIdx.x * blockDim.x + tid) >> 5; job < (3 * NN) / 16; job += nw) {
    int mbase = job * 16;
    v2f a[32];
    load_a32(vec + (size_t)(mbase + l16) * 128, hf, a);
    for (int ct = 0; ct < 32; ct += 2) {
      v8f u0 = splat8(0.f), u1 = splat8(0.f);
      gemm_pair(a, wl, 512, ct * 16, (ct + 1) * 16, l16, hf, u0, u1);
#pragma unroll
      for (int v = 0; v < 8; ++v) {
        size_t r = (size_t)(mbase + v + 8 * hf) * 512;
        P2[r + ct * 16 + l16] = u0[v];
        P2[r + (ct + 1) * 16 + l16] = u1[v];
      }
    }
  }
}

// ---------------- Kernel 4: fused edge attention/message.
// dk/dv GEMMs -> per-head attention (shfl reduce) -> vj -> x_agg atomics ->
// vj transpose via per-wave LDS -> s = silu(vj@Ws+bs) -> vec_msg -> vec_agg atomics.
__global__ __launch_bounds__(224) void k_edge_msg(
    const float* __restrict__ fij, const float* __restrict__ qkv,
    const float* __restrict__ vec,
    const int* __restrict__ eidx, const float* __restrict__ rij,
    const float* __restrict__ dij,
    const float* __restrict__ Wdk, const float* __restrict__ bdk,
    const float* __restrict__ Wdv, const float* __restrict__ bdv,
    const float* __restrict__ Ws, const float* __restrict__ bsb,
    float* __restrict__ xagg, float* __restrict__ vagg) {
  __shared__ v2f wl[64 * 512];          // Wdk | Wdv | Ws(256)  = 256 KB
  __shared__ float scr[7 * 16 * 132];   // per-wave vj transpose tiles
  int tid = threadIdx.x;
  load_w(wl, 512, 0, Wdk, 128, tid, blockDim.x);
  load_w(wl, 512, 128, Wdv, 128, tid, blockDim.x);
  load_w(wl, 512, 256, Ws, 256, tid, blockDim.x);
  __syncthreads();
  int lane = tid & 31, wave = tid >> 5, l16 = lane & 15, hf = lane >> 4;
  float* myscr = scr + wave * (16 * 132);
  const int* srcA = eidx;
  const int* dstA = eidx + EE;
  int nw = gridDim.x * 7;
  for (int job = blockIdx.x * 7 + wave; job < EE / 16; job += nw) {
    int ebase = job * 16;
    int eL = ebase + l16;
    int sL = srcA[eL], dL = dstA[eL];
    float rL = rij[eL];
    float d0L = dij[3 * eL], d1L = dij[3 * eL + 1], d2L = dij[3 * eL + 2];
    v2f a[32];
    load_a32_nt(fij + (size_t)eL * 128, hf, a);
    for (int head = 0; head < 8; ++head) {
      int c = head * 16 + l16;
      v8f cdk = splat8(bdk[c]);
      v8f cdv = splat8(bdv[c]);
      gemm_pair(a, wl, 512, head * 16, 128 + head * 16, l16, hf, cdk, cdv);
#pragma unroll
      for (int v = 0; v < 8; ++v) { cdk[v] = silu_f(cdk[v]); cdv[v] = silu_f(cdv[v]); }
#pragma unroll
      for (int v = 0; v < 8; ++v) {
        int sl = v + 8 * hf;
        int sv = __shfl(sL, sl, 32);
        int dv_ = __shfl(dL, sl, 32);
        float rv = __shfl(rL, sl, 32);
        float qv = qkv[(size_t)dv_ * 384 + c];
        float kv = qkv[(size_t)sv * 384 + 128 + c];
        float t = qv * kv * cdk[v];
        t += __shfl_xor(t, 1, 32);
        t += __shfl_xor(t, 2, 32);
        t += __shfl_xor(t, 4, 32);
        t += __shfl_xor(t, 8, 32);
        float ap = silu_f(t) * cutoff_f(rv);
        float vv = qkv[(size_t)sv * 384 + 256 + c];
        float vjv = vv * cdv[v] * ap;
        atomicAdd(xagg + (size_t)dv_ * 128 + c, vjv);
        myscr[sl * 132 + c] = vjv;  // padded, conflict-free
      }
    }
    // re-load vj tile as A fragments (same-wave LDS ops are in order)
#pragma unroll
    for (int s = 0; s < 32; ++s)
      a[s] = *(const v2f*)(myscr + l16 * 132 + 4 * s + 2 * hf);
    v8f s1f[8];
#pragma unroll
    for (int ct = 0; ct < 8; ct += 2) {
      v8f u0 = splat8(bsb[ct * 16 + l16]);
      v8f u1 = splat8(bsb[(ct + 1) * 16 + l16]);
      gemm_pair(a, wl, 512, 256 + ct * 16, 256 + (ct + 1) * 16, l16, hf, u0, u1);
#pragma unroll
      for (int v = 0; v < 8; ++v) { u0[v] = silu_f(u0[v]); u1[v] = silu_f(u1[v]); }
      s1f[ct] = u0; s1f[ct + 1] = u1;
    }
#pragma unroll
    for (int ct = 0; ct < 8; ct += 2) {
      v8f w0 = splat8(bsb[128 + ct * 16 + l16]);
      v8f w1 = splat8(bsb[128 + (ct + 1) * 16 + l16]);
      gemm_pair(a, wl, 512, 384 + ct * 16, 384 + (ct + 1) * 16, l16, hf, w0, w1);
#pragma unroll
      for (int v = 0; v < 8; ++v) { w0[v] = silu_f(w0[v]); w1[v] = silu_f(w1[v]); }
#pragma unroll
      for (int tt = 0; tt < 2; ++tt) {
        int c = (ct + tt) * 16 + l16;
        v8f s2v = tt ? w1 : w0;
        v8f s1v = s1f[ct + tt];
#pragma unroll
        for (int v = 0; v < 8; ++v) {
          int sl = v + 8 * hf;
          int sv = __shfl(sL, sl, 32);
          int dv_ = __shfl(dL, sl, 32);
          float dd0 = __shfl(d0L, sl, 32);
          float dd1 = __shfl(d1L, sl, 32);
          float dd2 = __shfl(d2L, sl, 32);
          size_t sb = (size_t)sv * 384, db = (size_t)dv_ * 384;
          float m0 = vec[sb + c] * s1v[v] + s2v[v] * dd0;
          float m1 = vec[sb + 128 + c] * s1v[v] + s2v[v] * dd1;
          float m2 = vec[sb + 256 + c] * s1v[v] + s2v[v] * dd2;
          atomicAdd(vagg + db + c, m0);
          atomicAdd(vagg + db + 128 + c, m1);
          atomicAdd(vagg + db + 256 + c, m2);
        }
      }
    }
  }
}

// ---------------- Kernel 5: edge update df_ij = f1*w_dot + f2*t_dot
// (rejection dots simplified: w1.w2 = g.h - (g.d)(h.d), d unit; -d invariant)
__global__ __launch_bounds__(256) void k_edge_df(
    const float* __restrict__ fij, const float* __restrict__ P2,
    const int* __restrict__ eidx, const float* __restrict__ dij,
    const float* __restrict__ Wf, const float* __restrict__ bf,
    float* __restrict__ dfout) {
  __shared__ v2f wl[64 * 256];
  int tid = threadIdx.x;
  load_w(wl, 256, 0, Wf, 256, tid, blockDim.x);
  __syncthreads();
  int lane = tid & 31, l16 = lane & 15, hf = lane >> 4;
  const int* srcA = eidx;
  const int* dstA = eidx + EE;
  int nw = (gridDim.x * blockDim.x) >> 5;
  for (int job = (blockIdx.x * blockDim.x + tid) >> 5; job < EE / 16; job += nw) {
    int ebase = job * 16;
    int eL = ebase + l16;
    int sL = srcA[eL], dL = dstA[eL];
    float d0L = dij[3 * eL], d1L = dij[3 * eL + 1], d2L = dij[3 * eL + 2];
    v2f a[32];
    load_a32_nt(fij + (size_t)eL * 128, hf, a);
    v8f f1f[8];
#pragma unroll
    for (int ct = 0; ct < 8; ct += 2) {
      v8f u0 = splat8(bf[ct * 16 + l16]);
      v8f u1 = splat8(bf[(ct + 1) * 16 + l16]);
      gemm_pair(a, wl, 256, ct * 16, (ct + 1) * 16, l16, hf, u0, u1);
#pragma unroll
      for (int v = 0; v < 8; ++v) { u0[v] = silu_f(u0[v]); u1[v] = silu_f(u1[v]); }
      f1f[ct] = u0; f1f[ct + 1] = u1;
    }
#pragma unroll
    for (int ct = 0; ct < 8; ct += 2) {
      v8f w0 = splat8(bf[128 + ct * 16 + l16]);
      v8f w1 = splat8(bf[128 + (ct + 1) * 16 + l16]);
      gemm_pair(a, wl, 256, 128 + ct * 16, 128 + (ct + 1) * 16, l16, hf, w0, w1);
#pragma unroll
      for (int v = 0; v < 8; ++v) { w0[v] = silu_f(w0[v]); w1[v] = silu_f(w1[v]); }
#pragma unroll
      for (int tt = 0; tt < 2; ++tt) {
        int c = (ct + tt) * 16 + l16;
        v8f f2v = tt ? w1 : w0;
        v8f f1v = f1f[ct + tt];
#pragma unroll
        for (int v = 0; v < 8; ++v) {
          int sl = v + 8 * hf;
          int sv = __shfl(sL, sl, 32);
          int dv_ = __shfl(dL, sl, 32);
          float dd0 = __shfl(d0L, sl, 32);
          float dd1 = __shfl(d1L, sl, 32);
          float dd2 = __shfl(d2L, sl, 32);
          size_t db = (size_t)dv_ * 3 * 512, sb = (size_t)sv * 3 * 512;
          float g0 = P2[db + c], g1 = P2[db + 512 + c], g2 = P2[db + 1024 + c];
          float h0 = P2[sb + 128 + c], h1 = P2[sb + 640 + c], h2 = P2[sb + 1152 + c];
          float p0 = P2[db + 256 + c], p1 = P2[db + 768 + c], p2 = P2[db + 1280 + c];
          float q0 = P2[db + 384 + c], q1 = P2[db + 896 + c], q2 = P2[db + 1408 + c];
          float wdot = (g0 * h0 + g1 * h1 + g2 * h2) -
                       (g0 * dd0 + g1 * dd1 + g2 * dd2) * (h0 * dd0 + h1 * dd1 + h2 * dd2);
          float tdot = (p0 * q0 + p1 * q1 + p2 * q2) -
                       (p0 * dd0 + p1 * dd1 + p2 * dd2) * (q0 * dd0 + q1 * dd1 + q2 * dd2);
          __builtin_nontemporal_store(f1v[v] * wdot + f2v[v] * tdot,
                                      dfout + (size_t)(ebase + sl) * 128 + c);
        }
      }
    }
  }
}

// ---------------- Kernel 6: o = x_agg@Wo+bo; dx = vec_dot*o2+o3; dvec = vec3*o1+vec_agg
__global__ __launch_bounds__(256) void k_nodeout(
    const float* __restrict__ xagg, const float* __restrict__ Wo,
    const float* __restrict__ bo, const float* __restrict__ vdotw,
    const float* __restrict__ vec3w, const float* __restrict__ vagg,
    float* __restrict__ dx, float* __restrict__ dvec) {
  __shared__ v2f wl[64 * 384];
  int tid = threadIdx.x;
  load_w(wl, 384, 0, Wo, 384, tid, blockDim.x);
  __syncthreads();
  int lane = tid & 31, l16 = lane & 15, hf = lane >> 4;
  int nw = (gridDim.x * blockDim.x) >> 5;
  for (int job = (blockIdx.x * blockDim.x + tid) >> 5; job < NN / 16; job += nw) {
    int mbase = job * 16;
    v2f a[32];
    load_a32(xagg + (size_t)(mbase + l16) * 128, hf, a);
    for (int ct = 0; ct < 8; ++ct) {
      int c = ct * 16 + l16;
      v8f o1 = splat8(bo[c]), o2 = splat8(bo[128 + c]), o3 = splat8(bo[256 + c]);
      gemm_pair(a, wl, 384, ct * 16, 128 + ct * 16, l16, hf, o1, o2);
      gemm_one(a, wl, 384, 256 + ct * 16, l16, hf, o3);
#pragma unroll
      for (int v = 0; v < 8; ++v) {
        int n = mbase + v + 8 * hf;
        size_t r = (size_t)n * 128 + c;
        dx[r] = vdotw[r] * o2[v] + o3[v];
        size_t r3 = (size_t)n * 384 + c;
        dvec[r3] = vec3w[r3] * o1[v] + vagg[r3];
        dvec[r3 + 128] = vec3w[r3 + 128] * o1[v] + vagg[r3 + 128];
        dvec[r3 + 256] = vec3w[r3 + 256] * o1[v] + vagg[r3 + 256];
      }
    }
  }
}

extern "C" void kernel_launch(void* const* d_in, const int* in_sizes, int n_in,
                              void* d_out, int out_size, void* d_ws, size_t ws_size,
                              hipStream_t stream) {
  (void)in_sizes; (void)n_in; (void)out_size; (void)ws_size;
  const float* x    = (const float*)d_in[0];
  const float* vec  = (const float*)d_in[1];
  const int*   eidx = (const int*)d_in[2];
  const float* rij  = (const float*)d_in[3];
  const float* fij  = (const float*)d_in[4];
  const float* dij  = (const float*)d_in[5];
  const float* Wq = (const float*)d_in[6];   const float* bq = (const float*)d_in[7];
  const float* Wk = (const float*)d_in[8];   const float* bk = (const float*)d_in[9];
  const float* Wv = (const float*)d_in[10];  const float* bv = (const float*)d_in[11];
  const float* Wdk = (const float*)d_in[12]; const float* bdk = (const float*)d_in[13];
  const float* Wdv = (const float*)d_in[14]; const float* bdv = (const float*)d_in[15];
  const float* Wvec = (const float*)d_in[16];
  const float* Ws = (const float*)d_in[17];  const float* bs = (const float*)d_in[18];
  const float* Wo = (const float*)d_in[19];  const float* bo = (const float*)d_in[20];
  const float* Wf = (const float*)d_in[21];  const float* bf = (const float*)d_in[22];
  const float* Wwsrc = (const float*)d_in[23];
  const float* Wwtrg = (const float*)d_in[24];
  const float* Wtsrc = (const float*)d_in[25];
  const float* Wttrg = (const float*)d_in[26];

  float* out   = (float*)d_out;
  float* dx    = out;
  float* dvec  = out + (size_t)NN * 128;
  float* dfout = dvec + (size_t)NN * 384;

  float* ws    = (float*)d_ws;
  float* qkv   = ws;                            // [N][384]
  float* vec3w = qkv + (size_t)NN * 384;        // [N*3][128]
  float* P2    = vec3w + (size_t)NN * 384;      // [N*3][512]
  float* vdotw = P2 + (size_t)NN * 3 * 512;     // [N][128]   (zeroed)
  float* xagg  = vdotw + (size_t)NN * 128;      // [N][128]   (zeroed)
  float* vagg  = xagg + (size_t)NN * 128;       // [N*3][128] (zeroed)

  hipMemsetAsync(vdotw, 0, (size_t)NN * (128 + 128 + 384) * sizeof(float), stream);

  k_qkv<<<79, 256, 0, stream>>>(x, Wq, bq, Wk, bk, Wv, bv, qkv);
  k_vecproj<<<235, 256, 0, stream>>>(vec, Wvec, vec3w, vdotw);
  k_vecproj2<<<235, 256, 0, stream>>>(vec, Wwtrg, Wwsrc, Wttrg, Wtsrc, P2);
  k_edge_msg<<<1024, 224, 0, stream>>>(fij, qkv, vec, eidx, rij, dij,
                                       Wdk, bdk, Wdv, bdv, Ws, bs, xagg, vagg);
  k_edge_df<<<1250, 256, 0, stream>>>(fij, P2, eidx, dij, Wf, bf, dfout);
  k_nodeout<<<79, 256, 0, stream>>>(xagg, Wo, bo, vdotw, vec3w, vagg, dx, dvec);
}